// TRCFModel_22136261443920
// MI455X (gfx1250) — compile-verified
//
#include <hip/hip_runtime.h>
#include <hip/hip_bf16.h>
#include <stdint.h>

// Problem constants (from the reference)
#define U_USERS 3000
#define S_SERV  1500
#define T_SLOTS 32
#define NBATCH  16384
#define KNN     50

#define UPB     2              // users per block (double-buffered TDM pipeline)
#define NPAD    3008           // 3000 padded to a multiple of 4 (b128 LDS loads)
#define NQUAD   (NPAD / 4)     // 752 uint4 quads

// ---------------------------------------------------------------------------
// CDNA5 feature detection (compile-time). TDM builtin arity differs between
// ROCm 7.2 (5 args) and amdgpu-toolchain/therock (6 args); the TDM header only
// ships with the latter, so use it as the arity discriminator. Round-1 compile
// confirmed the 5-arg form lowers to `tensor_load_to_lds` on this toolchain.
// ---------------------------------------------------------------------------
#if defined(__HIP_DEVICE_COMPILE__) && defined(__gfx1250__)
  #if __has_builtin(__builtin_amdgcn_tensor_load_to_lds) && \
      __has_builtin(__builtin_amdgcn_s_wait_tensorcnt)
    #define TRCF_TDM 1
  #endif
#endif

#if __has_include(<hip/amd_detail/amd_gfx1250_TDM.h>)
  #define TRCF_TDM_6ARGS 1
#endif

typedef __attribute__((ext_vector_type(4))) unsigned int tdm_u32x4;
typedef __attribute__((ext_vector_type(8))) int          tdm_i32x8;
typedef __attribute__((ext_vector_type(4))) int          tdm_i32x4;

// Monotone (order-preserving) float <-> u32 key mapping.
__device__ __forceinline__ uint32_t f2key(float f) {
  uint32_t b = __float_as_uint(f);
  return (b & 0x80000000u) ? ~b : (b | 0x80000000u);
}
__device__ __forceinline__ float key2f(uint32_t k) {
  uint32_t b = (k & 0x80000000u) ? (k & 0x7FFFFFFFu) : ~k;
  return __uint_as_float(b);
}

#if defined(TRCF_TDM)
// Issue one TDM DMA: 3000 contiguous f32 (one user_sim row) -> LDS.
__device__ __forceinline__ void tdm_load_row(const float* row, uint32_t lds) {
  const uint64_t ga = (uint64_t)(uintptr_t)row;

  tdm_u32x4 g0;
  g0[0] = 1u;                                              // count=1 (valid)
  g0[1] = lds;                                             // lds_addr (bytes)
  g0[2] = (uint32_t)ga;                                    // global_addr[31:0]
  g0[3] = ((uint32_t)(ga >> 32) & 0x01FFFFFFu)             // global_addr[56:32]
          | (2u << 30);                                    // type = 2 ("image")

  tdm_i32x8 g1;
  g1[0] = (int)(2u << 16);                                 // data_size = 4 B
  g1[1] = (int)(((uint32_t)U_USERS & 0xFFFFu) << 16);      // tensor_dim0[15:0]
  g1[2] = (int)((1u << 16) | ((U_USERS >> 16) & 0xFFFFu)); // tensor_dim1=1|dim0 hi
  g1[3] = (int)(((uint32_t)U_USERS & 0xFFFFu) << 16);      // tile_dim0 = 3000
  g1[4] = 1;                                               // tile_dim1 = 1
  g1[5] = U_USERS;                                         // tensor_dim0_stride lo
  g1[6] = (int)(((uint32_t)U_USERS & 0xFFFFu) << 16);      // dim1_stride lo16
  g1[7] = 0;

  tdm_i32x4 z4 = {0, 0, 0, 0};
#if defined(TRCF_TDM_6ARGS)
  tdm_i32x8 z8 = {0, 0, 0, 0, 0, 0, 0, 0};
  __builtin_amdgcn_tensor_load_to_lds(g0, g1, z4, z4, z8, 0);
#else
  __builtin_amdgcn_tensor_load_to_lds(g0, g1, z4, z4, 0);
#endif
}
#endif

// ---------------------------------------------------------------------------
// Phase 1: per-user top-50 of user_sim row. One wave32 per block, UPB users
// per block with double-buffered TDM loads: the DMA for user k+1 overlaps the
// 50 extraction passes for user k (TENSORcnt is in-order per wave, so
// s_wait_tensorcnt(1) proves load k is complete while k+1 is in flight).
// Extraction scans read LDS as uint4 (ds_load_b128) to cut DS issue count 4x.
// Tie-break: smaller index wins (matches jax.lax.top_k) via packed
// key64 = (monotone_key32 << 32) | ~idx.
// ---------------------------------------------------------------------------
__global__ __launch_bounds__(32) void trcf_topk(const float* __restrict__ user_sim,
                                                float* __restrict__ nb_val,
                                                int*   __restrict__ nb_idx) {
  __shared__ __align__(16) uint32_t keys[2][NPAD];   // 2 x 12032 B
  const int u0   = blockIdx.x * UPB;
  const int lane = threadIdx.x;

#if defined(TRCF_TDM)
  tdm_load_row(user_sim + (size_t)u0 * U_USERS, (uint32_t)(uintptr_t)&keys[0][0]);
#endif

  for (int k = 0; k < UPB; ++k) {
    const int u = u0 + k;
    uint32_t* buf = keys[k & 1];

#if defined(TRCF_TDM)
    if (k + 1 < UPB) {
      // Kick off next user's DMA into the other buffer, then wait only for
      // the current one (<=1 outstanding) so DMA overlaps extraction.
      tdm_load_row(user_sim + (size_t)(u + 1) * U_USERS,
                   (uint32_t)(uintptr_t)&keys[(k + 1) & 1][0]);
      __builtin_amdgcn_s_wait_tensorcnt(1);
    } else {
      __builtin_amdgcn_s_wait_tensorcnt(0);
    }
    __syncthreads();
    // Convert staged raw f32 bits to monotone keys in place; zero the pad.
    for (int q = lane; q < NQUAD; q += 32) {
      const int i0 = q * 4;
      uint4 raw = *reinterpret_cast<const uint4*>(&buf[i0]);
      uint4 kk;
      kk.x = (i0 + 0 < U_USERS) ? f2key(__uint_as_float(raw.x)) : 0u;
      kk.y = (i0 + 1 < U_USERS) ? f2key(__uint_as_float(raw.y)) : 0u;
      kk.z = (i0 + 2 < U_USERS) ? f2key(__uint_as_float(raw.z)) : 0u;
      kk.w = (i0 + 3 < U_USERS) ? f2key(__uint_as_float(raw.w)) : 0u;
      *reinterpret_cast<uint4*>(&buf[i0]) = kk;
    }
#else
    // Fallback staging: plain global loads, converting on the fly.
    const float* row = user_sim + (size_t)u * U_USERS;
    for (int q = lane; q < NQUAD; q += 32) {
      const int i0 = q * 4;
      uint4 kk;
      kk.x = (i0 + 0 < U_USERS) ? f2key(row[i0 + 0]) : 0u;
      kk.y = (i0 + 1 < U_USERS) ? f2key(row[i0 + 1]) : 0u;
      kk.z = (i0 + 2 < U_USERS) ? f2key(row[i0 + 2]) : 0u;
      kk.w = (i0 + 3 < U_USERS) ? f2key(row[i0 + 3]) : 0u;
      *reinterpret_cast<uint4*>(&buf[i0]) = kk;
    }
#endif
    __syncthreads();

    // 50 extraction passes: wave-wide argmax over packed (key, ~idx).
    for (int j = 0; j < KNN; ++j) {
      unsigned long long best = 0ull;
      for (int q = lane; q < NQUAD; q += 32) {
        const int i0 = q * 4;
        const uint4 kk = *reinterpret_cast<const uint4*>(&buf[i0]);
        unsigned long long c;
        c = ((unsigned long long)kk.x << 32) | (unsigned int)(~(i0 + 0));
        best = (c > best) ? c : best;
        c = ((unsigned long long)kk.y << 32) | (unsigned int)(~(i0 + 1));
        best = (c > best) ? c : best;
        c = ((unsigned long long)kk.z << 32) | (unsigned int)(~(i0 + 2));
        best = (c > best) ? c : best;
        c = ((unsigned long long)kk.w << 32) | (unsigned int)(~(i0 + 3));
        best = (c > best) ? c : best;
      }
      // Butterfly reduce across the wave32 (all lanes converge on the max).
      for (int off = 16; off > 0; off >>= 1) {
        unsigned long long o = __shfl_xor(best, off, 32);
        best = (o > best) ? o : best;
      }
      const int idx = (int)(~(unsigned int)(best & 0xFFFFFFFFull));
      if (lane == 0) {
        nb_idx[u * KNN + j] = idx;
        nb_val[u * KNN + j] = key2f((uint32_t)(best >> 32));
        buf[idx] = 0u;   // remove winner (0 is below every real key)
      }
      __syncthreads();
    }
  }
}

// ---------------------------------------------------------------------------
// Phase 2: per-batch-element prediction from the cached per-user top-50.
// base + (sum w*dev / sum w) with masking, clamped at 0.
// ---------------------------------------------------------------------------
__global__ __launch_bounds__(256) void trcf_predict(
    const float* __restrict__ qos,   const int* __restrict__ mask,
    const float* __restrict__ avg,   const int* __restrict__ tid,
    const int*   __restrict__ uid,   const int* __restrict__ sid,
    const float* __restrict__ nb_val,const int* __restrict__ nb_idx,
    float* __restrict__ out) {
  const int b = blockIdx.x * blockDim.x + threadIdx.x;
  if (b >= NBATCH) return;

  const int u = uid[b], s = sid[b], t = tid[b];
  const float base = avg[(size_t)u * S_SERV + s];
  const int*   ip = nb_idx + (size_t)u * KNN;
  const float* vp = nb_val + (size_t)u * KNN;

  // Issue CDNA5 prefetches for all 50 gathered lines up front
  // (lowers to global_prefetch_b8; hides the random-gather latency).
  for (int j = 0; j < KNN; ++j) {
    const int nb = ip[j];
    const size_t o3 = ((size_t)nb * S_SERV + s) * T_SLOTS + t;
    __builtin_prefetch(&qos[o3],  0, 0);
    __builtin_prefetch(&mask[o3], 0, 0);
  }

  float wsum = 0.0f, dsum = 0.0f;
  for (int j = 0; j < KNN; ++j) {
    const int nb = ip[j];
    const size_t o2 = (size_t)nb * S_SERV + s;
    const size_t o3 = o2 * T_SLOTS + t;
    if (mask[o3] != 0) {
      const float w = vp[j];
      wsum += w;
      dsum += w * (qos[o3] - avg[o2]);
    }
  }
  const float dev = (wsum > 0.0f) ? (dsum / wsum) : 0.0f;
  out[b] = fmaxf(base + dev, 0.0f);
}

// ---------------------------------------------------------------------------
// Launch: phase 1 over all 3000 users (16384 batch draws over 3000 users ->
// ~5.5x reuse, so per-user dedup is a big win), then phase 2 over the batch.
// Workspace: 3000*50 floats + 3000*50 ints = 1.2 MB, fully rewritten per call.
// ---------------------------------------------------------------------------
extern "C" void kernel_launch(void* const* d_in, const int* in_sizes, int n_in,
                              void* d_out, int out_size, void* d_ws, size_t ws_size,
                              hipStream_t stream) {
  (void)in_sizes; (void)n_in; (void)out_size; (void)ws_size;
  const float* qos  = (const float*)d_in[0];
  const int*   mask = (const int*)  d_in[1];
  const float* avg  = (const float*)d_in[2];
  const float* usim = (const float*)d_in[3];
  const int*   tid  = (const int*)  d_in[4];
  const int*   uid  = (const int*)  d_in[5];
  const int*   sid  = (const int*)  d_in[6];
  float* out = (float*)d_out;

  float* nbv = (float*)d_ws;
  int*   nbi = (int*)((char*)d_ws + sizeof(float) * (size_t)U_USERS * KNN);

  trcf_topk<<<U_USERS / UPB, 32, 0, stream>>>(usim, nbv, nbi);
  trcf_predict<<<(NBATCH + 255) / 256, 256, 0, stream>>>(
      qos, mask, avg, tid, uid, sid, nbv, nbi, out);
}